// MapEncoder_8229157339704
// MI455X (gfx1250) — compile-verified
//
#include <hip/hip_runtime.h>
#include <hip/hip_bf16.h>
#include <math.h>

typedef __attribute__((ext_vector_type(2))) float v2f;
typedef __attribute__((ext_vector_type(8))) float v8f;

#define NUM_HEADS 8
#define HEAD_DIM 32
#define F_CONT 16
#define D_EMB 240
#define D_MODEL 256          // = F_CONT + D_EMB = NUM_HEADS*HEAD_DIM
#define LEAKY 0.2f
#define XSTRIDE 264          // 256 + 8 pad: row stride not multiple of 64 banks

// ---------------------------------------------------------------------------
// Kernel 1: fused embedding lookup + concat + fp32 WMMA GEMM  proj = x @ W
// Block = 512 threads (16 waves). Each block computes a 16-row tile of proj;
// each wave owns one 16-column tile and chains 64 V_WMMA_F32_16X16X4_F32.
// ---------------------------------------------------------------------------
__global__ __launch_bounds__(512)
void embed_gemm_wmma(const float* __restrict__ cont,
                     const int* __restrict__ rid,
                     const int* __restrict__ lid,
                     const float* __restrict__ remb,
                     const float* __restrict__ lemb,
                     const float* __restrict__ W,
                     float* __restrict__ proj,
                     int N) {
    __shared__ float Xs[16 * XSTRIDE];
    const int tid  = threadIdx.x;
    const int row0 = blockIdx.x * 16;

    // Stage X tile [16 x 256] into LDS: 512 threads x 8 consecutive cols.
    {
        const int r  = tid >> 5;          // 0..15
        const int c0 = (tid & 31) * 8;    // 0..248
        int n  = row0 + r;
        int nc = (n < N) ? n : (N - 1);
        const int rI = rid[nc];
        const int lI = lid[nc];
        #pragma unroll
        for (int j = 0; j < 8; ++j) {
            const int c = c0 + j;
            float v;
            if (c < F_CONT) {
                v = cont[nc * F_CONT + c];
            } else {
                const int ce = c - F_CONT;
                v = remb[rI * D_EMB + ce] + lemb[lI * D_EMB + ce];
            }
            Xs[r * XSTRIDE + c] = v;
        }
    }
    __syncthreads();

    const int wave = tid >> 5;            // 0..15 -> column tile
    const int lane = tid & 31;
    const int half = lane >> 4;           // 0: K=0,1   1: K=2,3
    const int m    = lane & 15;
    const int col0 = wave * 16;

    v8f acc = {};
    for (int k = 0; k < D_MODEL; k += 4) {
        const int kk = k + 2 * half;
        // A (16x4, M x K): lane m holds row m, VGPR0=K(kk), VGPR1=K(kk+1)
        v2f a, b;
        a.x = Xs[m * XSTRIDE + kk];
        a.y = Xs[m * XSTRIDE + kk + 1];
        // B (4x16, K x N): lane m holds col col0+m, VGPR0=row kk, VGPR1=row kk+1
        b.x = W[kk * D_MODEL + col0 + m];
        b.y = W[(kk + 1) * D_MODEL + col0 + m];
        acc = __builtin_amdgcn_wmma_f32_16x16x4_f32(
            /*neg_a=*/false, a, /*neg_b=*/false, b,
            /*c_mod=*/(short)0, acc, /*reuse_a=*/false, /*reuse_b=*/false);
    }

    // D layout (16x16 f32): VGPR j -> row j + 8*half, col = col0 + (lane&15)
    // Uniform fast path: the whole 16-row tile is in range (always true when
    // N % 16 == 0) -> 8 unguarded stores, no per-element exec juggling.
    float* pbase = proj + (size_t)(row0 + 8 * half) * D_MODEL + col0 + m;
    if (row0 + 16 <= N) {
        #pragma unroll
        for (int j = 0; j < 8; ++j) {
            pbase[(size_t)j * D_MODEL] = acc[j];
        }
    } else {
        #pragma unroll
        for (int j = 0; j < 8; ++j) {
            if (row0 + 8 * half + j < N) pbase[(size_t)j * D_MODEL] = acc[j];
        }
    }
}

// ---------------------------------------------------------------------------
// Kernel 2: per-node per-head attention scores  s = <proj[n,h,:], a[h,:]>
// ---------------------------------------------------------------------------
__global__ void scores_kernel(const float* __restrict__ proj,
                              const float* __restrict__ a_src,
                              const float* __restrict__ a_trg,
                              float* __restrict__ s_src,
                              float* __restrict__ s_trg, int N) {
    const int idx = blockIdx.x * blockDim.x + threadIdx.x;
    if (idx >= N * NUM_HEADS) return;
    const int n = idx >> 3;
    const int h = idx & 7;
    const float* p  = proj + (size_t)n * D_MODEL + h * HEAD_DIM;
    const float* as = a_src + h * HEAD_DIM;
    const float* at = a_trg + h * HEAD_DIM;
    float ss = 0.f, st = 0.f;
    #pragma unroll
    for (int f = 0; f < HEAD_DIM; ++f) {
        const float v = p[f];
        ss += v * as[f];
        st += v * at[f];
    }
    s_src[idx] = ss;
    s_trg[idx] = st;
}

// ---------------------------------------------------------------------------
// Kernel 3: init accumulators (harness poisons buffers; must re-init per call)
// ---------------------------------------------------------------------------
__global__ void init_kernel(float* __restrict__ out,
                            unsigned int* __restrict__ emax_bits,
                            float* __restrict__ denom, int N) {
    const int idx = blockIdx.x * blockDim.x + threadIdx.x;
    if (idx < N * D_MODEL) out[idx] = 0.f;
    if (idx < N * NUM_HEADS) {
        emax_bits[idx] = 0xFF800000u;   // -inf
        denom[idx] = 0.f;
    }
}

// float atomic max via signed-max / unsigned-min bit trick (init must be -inf)
__device__ __forceinline__ void atomicMaxF32(float* addr, float value) {
    if (value >= 0.f) {
        atomicMax((int*)addr, __float_as_int(value));
    } else {
        atomicMin((unsigned int*)addr, __float_as_uint(value));
    }
}

__device__ __forceinline__ float leaky_logit(const float* s_src, const float* s_trg,
                                             int s, int t, int h) {
    float v = s_src[s * NUM_HEADS + h] + s_trg[t * NUM_HEADS + h];
    return (v > 0.f) ? v : LEAKY * v;
}

// ---------------------------------------------------------------------------
// Kernel 4: per-(edge,head) segment max of leaky-relu logits
// ---------------------------------------------------------------------------
__global__ void edge_max_kernel(const int* __restrict__ ei,
                                const float* __restrict__ s_src,
                                const float* __restrict__ s_trg,
                                float* __restrict__ emax, int E) {
    const int idx = blockIdx.x * blockDim.x + threadIdx.x;
    if (idx >= E * NUM_HEADS) return;
    const int e = idx >> 3, h = idx & 7;
    const int s = ei[e], t = ei[E + e];
    atomicMaxF32(emax + t * NUM_HEADS + h, leaky_logit(s_src, s_trg, s, t, h));
}

// ---------------------------------------------------------------------------
// Kernel 5: per-(edge,head) softmax denominator
// ---------------------------------------------------------------------------
__global__ void edge_denom_kernel(const int* __restrict__ ei,
                                  const float* __restrict__ s_src,
                                  const float* __restrict__ s_trg,
                                  const float* __restrict__ emax,
                                  float* __restrict__ denom, int E) {
    const int idx = blockIdx.x * blockDim.x + threadIdx.x;
    if (idx >= E * NUM_HEADS) return;
    const int e = idx >> 3, h = idx & 7;
    const int s = ei[e], t = ei[E + e];
    const float v = leaky_logit(s_src, s_trg, s, t, h);
    atomicAdd(denom + t * NUM_HEADS + h, expf(v - emax[t * NUM_HEADS + h]));
}

// ---------------------------------------------------------------------------
// Kernel 6: weighted message scatter-add. proj & out are L2-resident (102 MB
// combined << 192 MB L2) so the atomic storm stays on-chip.
// ---------------------------------------------------------------------------
__global__ void edge_agg_kernel(const int* __restrict__ ei,
                                const float* __restrict__ s_src,
                                const float* __restrict__ s_trg,
                                const float* __restrict__ emax,
                                const float* __restrict__ denom,
                                const float* __restrict__ proj,
                                float* __restrict__ out, int E) {
    const int idx = blockIdx.x * blockDim.x + threadIdx.x;
    if (idx >= E * NUM_HEADS) return;
    const int e = idx >> 3, h = idx & 7;
    const int s = ei[e], t = ei[E + e];
    const float v = leaky_logit(s_src, s_trg, s, t, h);
    const float alpha = expf(v - emax[t * NUM_HEADS + h]) /
                        (denom[t * NUM_HEADS + h] + 1e-16f);
    const float* p = proj + (size_t)s * D_MODEL + h * HEAD_DIM;
    float* o = out + (size_t)t * D_MODEL + h * HEAD_DIM;
    #pragma unroll
    for (int f = 0; f < HEAD_DIM; f += 4) {
        const float4 pv = *(const float4*)(p + f);
        atomicAdd(o + f + 0, pv.x * alpha);
        atomicAdd(o + f + 1, pv.y * alpha);
        atomicAdd(o + f + 2, pv.z * alpha);
        atomicAdd(o + f + 3, pv.w * alpha);
    }
}

// ---------------------------------------------------------------------------
// Kernel 7: out = relu(out + bias)
// ---------------------------------------------------------------------------
__global__ void finish_kernel(float* __restrict__ out,
                              const float* __restrict__ bias, int N) {
    const int idx = blockIdx.x * blockDim.x + threadIdx.x;
    if (idx >= N * D_MODEL) return;
    const float v = out[idx] + bias[idx & (D_MODEL - 1)];
    out[idx] = (v > 0.f) ? v : 0.f;
}

extern "C" void kernel_launch(void* const* d_in, const int* in_sizes, int n_in,
                              void* d_out, int out_size, void* d_ws, size_t ws_size,
                              hipStream_t stream) {
    const float* cont  = (const float*)d_in[0];
    const int*   rid   = (const int*)d_in[1];
    const int*   lid   = (const int*)d_in[2];
    const int*   ei    = (const int*)d_in[3];
    const float* remb  = (const float*)d_in[4];
    const float* lemb  = (const float*)d_in[5];
    const float* W     = (const float*)d_in[6];
    const float* a_src = (const float*)d_in[7];
    const float* a_trg = (const float*)d_in[8];
    const float* bias  = (const float*)d_in[9];
    float* out = (float*)d_out;

    const int N = in_sizes[1];
    const int E = in_sizes[3] / 2;

    // Workspace layout (all fp32): proj | s_src | s_trg | emax | denom
    char* ws = (char*)d_ws;
    float* proj  = (float*)ws;  ws += (size_t)N * D_MODEL * sizeof(float);
    float* ssrc  = (float*)ws;  ws += (size_t)N * NUM_HEADS * sizeof(float);
    float* strg  = (float*)ws;  ws += (size_t)N * NUM_HEADS * sizeof(float);
    float* emax  = (float*)ws;  ws += (size_t)N * NUM_HEADS * sizeof(float);
    float* denom = (float*)ws;

    const int tileBlocks = (N + 15) / 16;
    embed_gemm_wmma<<<tileBlocks, 512, 0, stream>>>(cont, rid, lid, remb, lemb, W, proj, N);

    const int scoreThreads = N * NUM_HEADS;
    scores_kernel<<<(scoreThreads + 255) / 256, 256, 0, stream>>>(proj, a_src, a_trg, ssrc, strg, N);

    const int initThreads = N * D_MODEL;
    init_kernel<<<(initThreads + 255) / 256, 256, 0, stream>>>(out, (unsigned int*)emax, denom, N);

    const int edgeThreads = E * NUM_HEADS;
    edge_max_kernel<<<(edgeThreads + 255) / 256, 256, 0, stream>>>(ei, ssrc, strg, emax, E);
    edge_denom_kernel<<<(edgeThreads + 255) / 256, 256, 0, stream>>>(ei, ssrc, strg, emax, denom, E);
    edge_agg_kernel<<<(edgeThreads + 255) / 256, 256, 0, stream>>>(ei, ssrc, strg, emax, denom, proj, out, E);

    finish_kernel<<<(initThreads + 255) / 256, 256, 0, stream>>>(out, bias, N);
}